// SimpleHGNLayer_15899968930383
// MI455X (gfx1250) — compile-verified
//
#include <hip/hip_runtime.h>

#define NPAPER 100000
#define NAUTH  50000
#define ECITES 200000
#define EWRITE 100000
#define EWB    100000
#define DIM    256
#define NH     8
#define SLOPE  0.2f
#define LNEPS  1e-5f

typedef __attribute__((ext_vector_type(2))) float v2f;
typedef __attribute__((ext_vector_type(8))) float v8f;

__device__ __forceinline__ float leaky_f(float x) { return x >= 0.f ? x : SLOPE * x; }

// float atomic max via monotonic int/uint trick (target must be initialized to -inf)
__device__ __forceinline__ void atomicMaxF(float* addr, float v) {
    if (v >= 0.f) atomicMax((int*)addr, __float_as_int(v));
    else          atomicMin((unsigned int*)addr, __float_as_uint(v));
}

// ---------------------------------------------------------------- fill
__global__ __launch_bounds__(256) void fill_f32(float* __restrict__ p, int n, float v) {
    int i = blockIdx.x * blockDim.x + threadIdx.x;
    int stride = gridDim.x * blockDim.x;
    for (; i < n; i += stride) p[i] = v;
}

// ---------------------------------------------------------------- GEMM: Y = X @ W^T
// X [nRowTiles*16, 256] row-major, W [256,256] row-major (so B = W^T, col-major over K).
// 512 threads = 16 waves; wave w owns output columns [16w,16w+16).
// B fragments (the wave's 16 W-rows) live in 64 v2f registers, reused across row tiles.
// A tiles double-buffered in LDS (stride 260: conflict-free v2f fragment reads, 16B-aligned
// b128 staging stores). Next tile's global loads are issued before the WMMA chain so the
// 64-deep v_wmma_f32_16x16x4_f32 accumulation hides the load latency.
__global__ __launch_bounds__(512) void gemm_xWT_wmma(const float* __restrict__ X,
                                                     const float* __restrict__ W,
                                                     float* __restrict__ Y,
                                                     int nRowTiles) {
    __shared__ float As[2][16 * 260];
    const int tid  = threadIdx.x;
    const int wave = tid >> 5;
    const int lane = tid & 31;
    const int j0   = wave * 16;        // output column base for this wave
    const int n    = lane & 15;        // column within tile
    const int kb   = (lane >> 4) * 2;  // K sub-offset per WMMA fragment layout

    // Preload B fragments: Breg[k4] = {W[j0+n][4k4+kb], W[j0+n][4k4+kb+1]}
    const float* wrow = W + (size_t)(j0 + n) * DIM + kb;
    v2f Breg[64];
#pragma unroll
    for (int k4 = 0; k4 < 64; ++k4)
        Breg[k4] = *(const v2f*)(wrow + k4 * 4);

    // Staging map: wave w stages row w of the 16x256 tile; lane covers 8 contiguous cols.
    const int srow = wave;
    const int scol = lane * 8;

    float4 s0, s1;
    int rt = blockIdx.x;
    if (rt < nRowTiles) {
        const float* xp = X + ((size_t)rt * 16 + srow) * DIM + scol;
        s0 = *(const float4*)xp;
        s1 = *(const float4*)(xp + 4);
    }

    int buf = 0;
    for (; rt < nRowTiles; rt += gridDim.x) {
        float* dst = &As[buf][srow * 260 + scol];
        *(float4*)dst       = s0;
        *(float4*)(dst + 4) = s1;
        __syncthreads();

        // prefetch next tile (overlaps with the WMMA chain below)
        int rtn = rt + gridDim.x;
        if (rtn < nRowTiles) {
            const float* xp = X + ((size_t)rtn * 16 + srow) * DIM + scol;
            s0 = *(const float4*)xp;
            s1 = *(const float4*)(xp + 4);
        }

        v8f acc = {};
        const float* Ab = &As[buf][n * 260 + kb];
#pragma unroll
        for (int k4 = 0; k4 < 64; ++k4) {
            acc = __builtin_amdgcn_wmma_f32_16x16x4_f32(
                false, *(const v2f*)(Ab + k4 * 4), false, Breg[k4], (short)0, acc, false, false);
        }

        // C layout: VGPR v holds (M = v + 8*(lane>=16), N = lane&15)
        float* Yt = Y + (size_t)rt * 16 * DIM + j0 + n;
        const int mbase = (lane >> 4) * 8;
#pragma unroll
        for (int v = 0; v < 8; ++v)
            Yt[(size_t)(mbase + v) * DIM] = acc[v];

        buf ^= 1;
    }
}

// ---------------------------------------------------------------- per-node attention dots
// wave per row: el[r,h] = sum_k H[r,h*32+k]*attn_l[h*32+k]; same for er.
__global__ __launch_bounds__(256) void node_scores(const float* __restrict__ H,
                                                   const float* __restrict__ al,
                                                   const float* __restrict__ ar,
                                                   float* __restrict__ el,
                                                   float* __restrict__ er, int nrows) {
    int wid  = (blockIdx.x * blockDim.x + threadIdx.x) >> 5;
    int lane = threadIdx.x & 31;
    if (wid >= nrows) return;
    const float* row = H + (size_t)wid * DIM;
    float pl[NH], pr[NH];
#pragma unroll
    for (int i = 0; i < NH; ++i) {
        float x = row[i * 32 + lane];
        pl[i] = x * al[i * 32 + lane];
        pr[i] = x * ar[i * 32 + lane];
    }
#pragma unroll
    for (int off = 16; off > 0; off >>= 1) {
#pragma unroll
        for (int i = 0; i < NH; ++i) {
            pl[i] += __shfl_xor(pl[i], off, 32);
            pr[i] += __shfl_xor(pr[i], off, 32);
        }
    }
    if (lane == 0) {
#pragma unroll
        for (int i = 0; i < NH; ++i) {
            el[wid * NH + i] = pl[i];
            er[wid * NH + i] = pr[i];
        }
    }
}

// ---------------------------------------------------------------- edge-type embedding term
// ee[r,h] = sum_dk ((edge_emb[r] @ fce_W^T)[h*32+dk] * attn_e[h*32+dk]); r = blockIdx.x in 0..2
__global__ __launch_bounds__(256) void edge_type_scores(const float* __restrict__ edge_emb,
                                                        const float* __restrict__ fceW,
                                                        const float* __restrict__ attn_e,
                                                        float* __restrict__ ee) {
    __shared__ float proj[DIM];
    int r = blockIdx.x, j = threadIdx.x;
    float s = 0.f;
    for (int k = 0; k < DIM; ++k) s += edge_emb[r * DIM + k] * fceW[j * DIM + k];
    proj[j] = s * attn_e[j];
    __syncthreads();
    if (j < NH) {
        float t = 0.f;
        for (int k = 0; k < 32; ++k) t += proj[j * 32 + k];
        ee[r * NH + j] = t;
    }
}

// ---------------------------------------------------------------- edge logits + segment max
__global__ __launch_bounds__(256) void edge_logits(const int* __restrict__ src,
                                                   const int* __restrict__ dst,
                                                   const float* __restrict__ el,
                                                   const float* __restrict__ er,
                                                   const float* __restrict__ ee,
                                                   float* __restrict__ sc,
                                                   float* __restrict__ mdst, int ne) {
    int e = blockIdx.x * blockDim.x + threadIdx.x;
    if (e >= ne) return;
    int s = src[e], d = dst[e];
#pragma unroll
    for (int h = 0; h < NH; ++h) {
        float v = leaky_f(el[s * NH + h] + er[d * NH + h] + ee[h]);
        sc[(size_t)e * NH + h] = v;
        atomicMaxF(&mdst[d * NH + h], v);
    }
}

// ---------------------------------------------------------------- exp(s - max) + segment sum
__global__ __launch_bounds__(256) void edge_exp(const int* __restrict__ dst,
                                                float* __restrict__ sc,
                                                const float* __restrict__ mdst,
                                                float* __restrict__ zdst, int ne) {
    int e = blockIdx.x * blockDim.x + threadIdx.x;
    if (e >= ne) return;
    int d = dst[e];
#pragma unroll
    for (int h = 0; h < NH; ++h) {
        float ex = __expf(sc[(size_t)e * NH + h] - mdst[d * NH + h]);
        sc[(size_t)e * NH + h] = ex;
        atomicAdd(&zdst[d * NH + h], ex);
    }
}

// ---------------------------------------------------------------- weighted scatter-sum
// wave per edge; lane covers cols {i*32+lane}, head = i; coef broadcast via shfl.
__global__ __launch_bounds__(256) void edge_aggregate(const int* __restrict__ src,
                                                      const int* __restrict__ dst,
                                                      const float* __restrict__ sc,
                                                      const float* __restrict__ zdst,
                                                      const float* __restrict__ Hsrc,
                                                      float* __restrict__ out, int ne) {
    int wid  = (blockIdx.x * blockDim.x + threadIdx.x) >> 5;
    int lane = threadIdx.x & 31;
    if (wid >= ne) return;
    int s = src[wid], d = dst[wid];
    float c = 0.f;
    if (lane < NH) c = sc[(size_t)wid * NH + lane] / zdst[d * NH + lane];
    const float* hrow = Hsrc + (size_t)s * DIM;
    float* orow = out + (size_t)d * DIM;
#pragma unroll
    for (int i = 0; i < NH; ++i) {
        float coef = __shfl(c, i, 32);
        atomicAdd(&orow[i * 32 + lane], hrow[i * 32 + lane] * coef);
    }
}

// ---------------------------------------------------------------- residual + leaky + LayerNorm (in place)
__global__ __launch_bounds__(256) void finalize_ln(float* __restrict__ out,
                                                   const float* __restrict__ Hproj,
                                                   const float* __restrict__ gamma,
                                                   const float* __restrict__ beta, int nrows) {
    int wid  = (blockIdx.x * blockDim.x + threadIdx.x) >> 5;
    int lane = threadIdx.x & 31;
    if (wid >= nrows) return;
    float* orow = out + (size_t)wid * DIM;
    const float* hr = Hproj + (size_t)wid * DIM;
    float x[NH];
    float sum = 0.f;
#pragma unroll
    for (int i = 0; i < NH; ++i) {
        float v = leaky_f(orow[i * 32 + lane] + hr[i * 32 + lane]);
        x[i] = v;
        sum += v;
    }
#pragma unroll
    for (int off = 16; off > 0; off >>= 1) sum += __shfl_xor(sum, off, 32);
    float mu = sum * (1.f / DIM);
    float var = 0.f;
#pragma unroll
    for (int i = 0; i < NH; ++i) {
        float dlt = x[i] - mu;
        var += dlt * dlt;
    }
#pragma unroll
    for (int off = 16; off > 0; off >>= 1) var += __shfl_xor(var, off, 32);
    float rinv = rsqrtf(var * (1.f / DIM) + LNEPS);
#pragma unroll
    for (int i = 0; i < NH; ++i)
        orow[i * 32 + lane] = (x[i] - mu) * rinv * gamma[i * 32 + lane] + beta[i * 32 + lane];
}

// ================================================================ launch
extern "C" void kernel_launch(void* const* d_in, const int* in_sizes, int n_in,
                              void* d_out, int out_size, void* d_ws, size_t ws_size,
                              hipStream_t stream) {
    (void)in_sizes; (void)n_in; (void)out_size; (void)ws_size;
    const float* h_paper  = (const float*)d_in[0];
    const float* h_author = (const float*)d_in[1];
    const float* fc_W     = (const float*)d_in[2];
    const float* fce_W    = (const float*)d_in[3];
    const float* edge_emb = (const float*)d_in[4];
    const float* attn_l   = (const float*)d_in[5];
    const float* attn_r   = (const float*)d_in[6];
    const float* attn_e   = (const float*)d_in[7];
    const float* gamma    = (const float*)d_in[8];
    const float* beta     = (const float*)d_in[9];
    const int* cites_src  = (const int*)d_in[10];
    const int* cites_dst  = (const int*)d_in[11];
    const int* writes_src = (const int*)d_in[12];
    const int* writes_dst = (const int*)d_in[13];
    const int* wb_src     = (const int*)d_in[14];
    const int* wb_dst     = (const int*)d_in[15];
    float* out = (float*)d_out;

    char* ws = (char*)d_ws;
    size_t off = 0;
    auto take = [&](size_t nfloats) -> float* {
        float* p = (float*)(ws + off);
        off += ((nfloats * sizeof(float) + 255) / 256) * 256;
        return p;
    };
    float* hp   = take((size_t)NPAPER * DIM);
    float* ha   = take((size_t)NAUTH * DIM);
    float* el_p = take((size_t)NPAPER * NH);
    float* er_p = take((size_t)NPAPER * NH);
    float* el_a = take((size_t)NAUTH * NH);
    float* er_a = take((size_t)NAUTH * NH);
    float* m_p  = take((size_t)NPAPER * NH);
    float* z_p  = take((size_t)NPAPER * NH);
    float* m_a  = take((size_t)NAUTH * NH);
    float* z_a  = take((size_t)NAUTH * NH);
    float* ee   = take(3 * NH);
    float* s_c  = take((size_t)ECITES * NH);
    float* s_w  = take((size_t)EWRITE * NH);
    float* s_b  = take((size_t)EWB * NH);

    const float NEG_INF = -__builtin_huge_valf();

    // ---- init (deterministic every call)
    fill_f32<<<2048, 256, 0, stream>>>(out, (NPAPER + NAUTH) * DIM, 0.f);
    fill_f32<<<512, 256, 0, stream>>>(m_p, NPAPER * NH, NEG_INF);
    fill_f32<<<512, 256, 0, stream>>>(z_p, NPAPER * NH, 0.f);
    fill_f32<<<256, 256, 0, stream>>>(m_a, NAUTH * NH, NEG_INF);
    fill_f32<<<256, 256, 0, stream>>>(z_a, NAUTH * NH, 0.f);

    // ---- projections (WMMA f32)
    int tiles_p = NPAPER / 16, tiles_a = NAUTH / 16;
    int gx_p = tiles_p < 2048 ? tiles_p : 2048;
    int gx_a = tiles_a < 2048 ? tiles_a : 2048;
    gemm_xWT_wmma<<<gx_p, 512, 0, stream>>>(h_paper, fc_W, hp, tiles_p);
    gemm_xWT_wmma<<<gx_a, 512, 0, stream>>>(h_author, fc_W, ha, tiles_a);

    // ---- edge-type embedding scores
    edge_type_scores<<<3, 256, 0, stream>>>(edge_emb, fce_W, attn_e, ee);

    // ---- per-node el/er
    node_scores<<<(NPAPER + 7) / 8, 256, 0, stream>>>(hp, attn_l, attn_r, el_p, er_p, NPAPER);
    node_scores<<<(NAUTH + 7) / 8, 256, 0, stream>>>(ha, attn_l, attn_r, el_a, er_a, NAUTH);

    // ---- edge logits + segment max (cross-relation softmax over shared dst)
    edge_logits<<<(ECITES + 255) / 256, 256, 0, stream>>>(cites_src, cites_dst, el_p, er_p, ee + 0, s_c, m_p, ECITES);
    edge_logits<<<(EWRITE + 255) / 256, 256, 0, stream>>>(writes_src, writes_dst, el_a, er_p, ee + NH, s_w, m_p, EWRITE);
    edge_logits<<<(EWB + 255) / 256, 256, 0, stream>>>(wb_src, wb_dst, el_p, er_a, ee + 2 * NH, s_b, m_a, EWB);

    // ---- exp + segment sum
    edge_exp<<<(ECITES + 255) / 256, 256, 0, stream>>>(cites_dst, s_c, m_p, z_p, ECITES);
    edge_exp<<<(EWRITE + 255) / 256, 256, 0, stream>>>(writes_dst, s_w, m_p, z_p, EWRITE);
    edge_exp<<<(EWB + 255) / 256, 256, 0, stream>>>(wb_dst, s_b, m_a, z_a, EWB);

    // ---- weighted scatter aggregation directly into d_out
    edge_aggregate<<<(ECITES + 7) / 8, 256, 0, stream>>>(cites_src, cites_dst, s_c, z_p, hp, out, ECITES);
    edge_aggregate<<<(EWRITE + 7) / 8, 256, 0, stream>>>(writes_src, writes_dst, s_w, z_p, ha, out, EWRITE);
    edge_aggregate<<<(EWB + 7) / 8, 256, 0, stream>>>(wb_src, wb_dst, s_b, z_a, hp, out + (size_t)NPAPER * DIM, EWB);

    // ---- residual + leaky + LayerNorm (in place on d_out)
    finalize_ln<<<(NPAPER + 7) / 8, 256, 0, stream>>>(out, hp, gamma, beta, NPAPER);
    finalize_ln<<<(NAUTH + 7) / 8, 256, 0, stream>>>(out + (size_t)NPAPER * DIM, ha, gamma, beta, NAUTH);
}